// SequencerRank_44117904065324
// MI455X (gfx1250) — compile-verified
//
#include <hip/hip_runtime.h>

// Problem constants from the reference (layout-critical ones hardcoded).
#define R_RANKS 4
#define F_FEAT  64
#define RF      (R_RANKS * F_FEAT)   // 256

typedef __attribute__((ext_vector_type(2))) float v2f;
typedef __attribute__((ext_vector_type(8))) float v8f;

// ---------------------------------------------------------------------------
// Phase 1: he_feat[c, :] += x[rows[i], :] * vals[i]   (one wave per nnz)
// x row and he_feat row are contiguous 64-float vectors -> coalesced, 2/lane.
// ---------------------------------------------------------------------------
__global__ void k_build_hefeat(const float* __restrict__ x,
                               const float* __restrict__ vals,
                               const int* __restrict__ rows,
                               const int* __restrict__ cols,
                               float* __restrict__ he_feat,
                               int nnz) {
  const int lane   = threadIdx.x & 31;
  const int wave   = blockIdx.x * (blockDim.x >> 5) + (threadIdx.x >> 5);
  const int nwaves = gridDim.x * (blockDim.x >> 5);
  for (int i = wave; i < nnz; i += nwaves) {
    const int   r = rows[i];
    const int   c = cols[i];
    const float v = vals[i];
    const float* xr = x + (long)r * F_FEAT + lane;
    float*       hf = he_feat + (long)c * F_FEAT + lane;
    atomicAdd(hf + 0,  xr[0]  * v);   // immediate-offset atomics
    atomicAdd(hf + 32, xr[32] * v);
  }
}

// ---------------------------------------------------------------------------
// Phase 2+3: for each nnz i:
//   out[rows[i], r, f] += vals[i] * rank_masks[r, he_idxs[cols[i]]] * he_feat[cols[i], f]
// The 4x64 outer product is one V_WMMA_F32_16X16X4_F32:
//   D[i16, j] = sum_k A[i16,k] * B[k,j]
//   A[i16,k] = (k == i16>>2) ? m[i16&3] : 0      (m already scaled by vals[i])
//   B[k, j]  = he_feat[c, 16*k + j]
//   => D[i16,j] = m[i16&3] * hf[16*(i16>>2) + j]  == out tile (rank=i16&3, f=16*(i16>>2)+j)
// A-layout (ISA 7.12.2, 32-bit A 16x4): lanes 0-15 hold K=0,1; lanes 16-31 K=2,3; M=lane%16.
// B 4x16: lanes 0-15 hold N=lane with K=0(v0),1(v1); lanes 16-31 N=lane-16, K=2,3.
// D-layout: VGPR v, lanes 0-15 -> M=v, N=lane; lanes 16-31 -> M=v+8, N=lane-16.
//
// Scatter targets relative to base = out + row*256 + j (+32 for lanes 16-31):
//   vg -> element offset (vg&3)*64 + (vg>>2)*16  = {0,64,128,192,16,80,144,208}
// All compile-time constants -> fold into global_atomic IOFFSET (no per-iter
// 64-bit address adds).  Wave-uniform loop => EXEC all-1s at every WMMA.
// ---------------------------------------------------------------------------
__global__ void k_wmma_scatter(const float* __restrict__ he_feat,
                               const float* __restrict__ rank_masks,
                               const float* __restrict__ vals,
                               const int* __restrict__ he_idxs,
                               const int* __restrict__ rows,
                               const int* __restrict__ cols,
                               float* __restrict__ out,
                               int nnz, int E) {
  const int lane   = threadIdx.x & 31;
  const int wave   = blockIdx.x * (blockDim.x >> 5) + (threadIdx.x >> 5);
  const int nwaves = gridDim.x * (blockDim.x >> 5);

  const int M    = (lane < 16) ? lane : (lane - 16);  // A/D matrix row this lane serves
  const int kx   = (lane < 16) ? 0 : 2;               // K indices this lane holds (kx, kx+1)
  const int j    = lane & 15;                         // B/D column this lane serves
  const int half = (lane < 16) ? 0 : 32;              // upper half-wave feature shift

  for (int i = wave; i < nnz; i += nwaves) {
    const int   row = rows[i];
    const int   c   = cols[i];
    const float v   = vals[i];
    const int   he  = he_idxs[c];

    // m value needed by this lane's A rows: m[M&3] * val
    const float mv = rank_masks[(long)(M & 3) * E + he] * v;

    v2f a;
    a.x = ((M >> 2) == kx)     ? mv : 0.0f;   // A[M, kx]
    a.y = ((M >> 2) == kx + 1) ? mv : 0.0f;   // A[M, kx+1]

    // B[kx, j] and B[kx+1, j] from one base pointer, +0 / +16 elements.
    const float* hfp = he_feat + (long)c * F_FEAT + j + half;
    v2f b;
    b.x = hfp[0];
    b.y = hfp[16];

    v8f c0 = {};
    v8f d = __builtin_amdgcn_wmma_f32_16x16x4_f32(
        /*neg_a=*/false, a, /*neg_b=*/false, b,
        /*c_mod=*/(short)0, c0, /*reuse_a=*/false, /*reuse_b=*/false);

    // One base address; 8 atomics with constant element offsets -> IOFFSET.
    float* o = out + (long)row * RF + j + half;
    atomicAdd(o + 0,   d[0]);   // rank 0, fblock 0
    atomicAdd(o + 64,  d[1]);   // rank 1, fblock 0
    atomicAdd(o + 128, d[2]);   // rank 2, fblock 0
    atomicAdd(o + 192, d[3]);   // rank 3, fblock 0
    atomicAdd(o + 16,  d[4]);   // rank 0, fblock 1
    atomicAdd(o + 80,  d[5]);   // rank 1, fblock 1
    atomicAdd(o + 144, d[6]);   // rank 2, fblock 1
    atomicAdd(o + 208, d[7]);   // rank 3, fblock 1
  }
}

// ---------------------------------------------------------------------------
// Phase 4: out[:, R-1, :] = x   (plain overwrite, matches .at[:, -1, :].set(x))
// ---------------------------------------------------------------------------
__global__ void k_set_last_rank(const float* __restrict__ x,
                                float* __restrict__ out,
                                long n_elems /* = N*F */) {
  long idx = (long)blockIdx.x * blockDim.x + threadIdx.x;
  if (idx < n_elems) {
    long n = idx >> 6;        // F = 64
    int  f = (int)(idx & 63);
    out[n * RF + (R_RANKS - 1) * F_FEAT + f] = x[idx];
  }
}

extern "C" void kernel_launch(void* const* d_in, const int* in_sizes, int n_in,
                              void* d_out, int out_size, void* d_ws, size_t ws_size,
                              hipStream_t stream) {
  const float* x          = (const float*)d_in[0];  // (N, 64)
  const float* rank_masks = (const float*)d_in[1];  // (4, E)
  const float* vals       = (const float*)d_in[2];  // (NNZ,)
  const int*   he_idxs    = (const int*)d_in[3];    // (E,)
  const int*   rows       = (const int*)d_in[4];    // (NNZ,)
  const int*   cols       = (const int*)d_in[5];    // (NNZ,)
  float*       out        = (float*)d_out;          // (N, 4, 64)
  float*       he_feat    = (float*)d_ws;           // (E, 64) scratch

  const int  nnz = in_sizes[2];
  const int  E   = in_sizes[3];
  const long NF  = (long)in_sizes[0];               // N*F

  // Zero accumulation targets (graph-capture-safe async memsets).
  hipMemsetAsync(out, 0, (size_t)out_size * sizeof(float), stream);
  hipMemsetAsync(he_feat, 0, (size_t)E * F_FEAT * sizeof(float), stream);

  const int threads = 256;                 // 8 wave32 per block
  const int wpb     = threads / 32;
  int blocks = (nnz + wpb - 1) / wpb;      // one wave per nnz (loops if capped)
  if (blocks > 131072) blocks = 131072;

  k_build_hefeat<<<blocks, threads, 0, stream>>>(x, vals, rows, cols, he_feat, nnz);
  k_wmma_scatter<<<blocks, threads, 0, stream>>>(he_feat, rank_masks, vals,
                                                 he_idxs, rows, cols, out, nnz, E);

  int b3 = (int)((NF + threads - 1) / threads);
  k_set_last_rank<<<b3, threads, 0, stream>>>(x, out, NF);
}